// PolyhedronModel_87213605912803
// MI455X (gfx1250) — compile-verified
//
#include <hip/hip_runtime.h>
#include <stdint.h>

#define NFEAT 128
#define FOUT  131   // 128 features + 3 pos deltas
#define NBUF  4     // async LDS ring buffers per wave (prefetch distance 3)

// ---------------------------------------------------------------- phase 0: zero counters
__global__ void pp_zero(unsigned* __restrict__ counts, int n) {
    int i = blockIdx.x * blockDim.x + threadIdx.x;
    if (i < n) counts[i] = 0u;
}

// ---------------------------------------------------------------- phase 1: histogram of dst
__global__ void pp_count(const long long* __restrict__ ei, unsigned* __restrict__ counts, int E) {
    int e = blockIdx.x * blockDim.x + threadIdx.x;
    if (e < E) {
        int dst = (int)ei[(size_t)E + e];  // row 1 of edge_index
        atomicAdd(&counts[dst], 1u);
    }
}

// ---------------------------------------------------------------- phase 2: exclusive scan (1 block)
__global__ void pp_scan(const unsigned* __restrict__ counts,
                        unsigned* __restrict__ offsets,
                        unsigned* __restrict__ cursor, int n) {
    __shared__ unsigned tmp[1024];
    __shared__ unsigned carry;
    const int t = threadIdx.x;
    if (t == 0) carry = 0u;
    __syncthreads();
    for (int base = 0; base < n; base += 1024) {
        const int i = base + t;
        const unsigned c = (i < n) ? counts[i] : 0u;
        tmp[t] = c;
        __syncthreads();
        // Hillis-Steele inclusive scan over 1024 threads
        for (int off = 1; off < 1024; off <<= 1) {
            unsigned v = (t >= off) ? tmp[t - off] : 0u;
            __syncthreads();
            tmp[t] += v;
            __syncthreads();
        }
        const unsigned excl = carry + tmp[t] - c;
        if (i < n) { offsets[i] = excl; cursor[i] = excl; }
        __syncthreads();                    // everyone read old carry
        if (t == 1023) carry += tmp[1023];  // add chunk total
        __syncthreads();
    }
    if (t == 0) offsets[n] = carry;
}

// ---------------------------------------------------------------- phase 3: scatter src ids into CSR
__global__ void pp_scatter(const long long* __restrict__ ei,
                           unsigned* __restrict__ cursor,
                           unsigned* __restrict__ bucket, int E) {
    int e = blockIdx.x * blockDim.x + threadIdx.x;
    if (e < E) {
        int src = (int)ei[e];
        int dst = (int)ei[(size_t)E + e];
        unsigned p = atomicAdd(&cursor[dst], 1u);
        bucket[p] = (unsigned)src;
    }
}

// ---------------------------------------------------------------- phase 4: per-node wave32 max-reduce
// One wave (32 lanes) per destination node; lane l owns features [4l, 4l+3].
// x rows (512 B) stream through a 4-deep LDS ring via global_load_async_to_lds_b128
// (ASYNCcnt), keeping 3 rows in flight to cover L2 gather latency.
__global__ void pn_reduce(const float* __restrict__ x,
                          const float* __restrict__ pos,
                          const unsigned* __restrict__ offsets,
                          const unsigned* __restrict__ bucket,
                          float* __restrict__ out, int n) {
    __shared__ __align__(16) float lds[8][NBUF][NFEAT];  // 8 waves x 4 bufs x 512B = 16KB

    const int lane = threadIdx.x & 31;
    const int wave = threadIdx.x >> 5;
    const int node = blockIdx.x * (blockDim.x >> 5) + wave;
    if (node >= n) return;  // wave-uniform: EXEC stays all-ones within live waves

    const unsigned beg = offsets[node];
    const unsigned deg = offsets[node + 1] - beg;
    const size_t obase = (size_t)node * FOUT;

    if (deg == 0u) {  // empty segment: reference fills with 0 (isfinite mask)
        out[obase + lane * 4 + 0] = 0.0f;
        out[obase + lane * 4 + 1] = 0.0f;
        out[obase + lane * 4 + 2] = 0.0f;
        out[obase + lane * 4 + 3] = 0.0f;
        if (lane < 3) out[obase + NFEAT + lane] = 0.0f;
        return;
    }

    const float pi = (lane < 3) ? pos[(size_t)node * 3 + lane] : 0.0f;
    const float ninf = -__builtin_inff();
    float a0 = ninf, a1 = ninf, a2 = ninf, a3 = ninf, pacc = ninf;

    // LDS byte offset of this lane's 16B slot in ring buffer 0
    // (low 32 bits of a generic pointer to __shared__ == LDS offset on AMDGPU).
    const unsigned ldsb = (unsigned)(size_t)&lds[wave][0][lane << 2];

    // ---- prologue: fill the pipeline (up to 3 rows in flight) ----
    const unsigned npre = (deg < 3u) ? deg : 3u;
    for (unsigned j = 0; j < npre; ++j) {
        const unsigned s = bucket[beg + j];
        asm volatile("global_load_async_to_lds_b128 %0, %1, off"
                     :: "v"(ldsb + ((j & (NBUF - 1u)) << 9)),
                        "v"(x + (size_t)s * NFEAT + (lane << 2)) : "memory");
    }

    for (unsigned k = 0; k < deg; ++k) {
        const unsigned srcK = bucket[beg + k];  // L0/L2-cached; needed for pos delta
        if (k + 3u < deg) {
            const unsigned s = bucket[beg + k + 3u];
            asm volatile("global_load_async_to_lds_b128 %0, %1, off"
                         :: "v"(ldsb + (((k + 3u) & (NBUF - 1u)) << 9)),
                            "v"(x + (size_t)s * NFEAT + (lane << 2)) : "memory");
            asm volatile("s_wait_asynccnt 3" ::: "memory");  // row k landed, 3 in flight
        } else {
            const unsigned rem = deg - 1u - k;  // 0..2 loads still wanted in flight
            if (rem >= 2u)      asm volatile("s_wait_asynccnt 2" ::: "memory");
            else if (rem == 1u) asm volatile("s_wait_asynccnt 1" ::: "memory");
            else                asm volatile("s_wait_asynccnt 0" ::: "memory");
        }

        const float4 v = *(const float4*)&lds[wave][k & (NBUF - 1u)][lane << 2];  // ds_load_b128
        // Ensure the DS read has delivered to VGPRs before any later async DMA may
        // overwrite this ring slot (ASYNCcnt and DScnt are independent counters).
        asm volatile("s_wait_dscnt 0" ::: "memory");

        a0 = fmaxf(a0, v.x); a1 = fmaxf(a1, v.y);
        a2 = fmaxf(a2, v.z); a3 = fmaxf(a3, v.w);
        if (lane < 3) pacc = fmaxf(pacc, pos[(size_t)srcK * 3 + lane] - pi);
    }

    // out row stride is 131 floats (524B) -> rows only 4B-aligned: scalar stores.
    out[obase + lane * 4 + 0] = a0;
    out[obase + lane * 4 + 1] = a1;
    out[obase + lane * 4 + 2] = a2;
    out[obase + lane * 4 + 3] = a3;
    if (lane < 3) out[obase + NFEAT + lane] = pacc;
}

// ----------------------------------------------------------------
extern "C" void kernel_launch(void* const* d_in, const int* in_sizes, int n_in,
                              void* d_out, int out_size, void* d_ws, size_t ws_size,
                              hipStream_t stream) {
    const float*     x   = (const float*)d_in[0];        // [N,128] f32
    const float*     pos = (const float*)d_in[1];        // [N,3]   f32
    const long long* ei  = (const long long*)d_in[2];    // [2,E]   i64
    float*           out = (float*)d_out;                // [N,131] f32

    const int N = in_sizes[0] / NFEAT;
    const int E = in_sizes[2] / 2;

    // workspace layout: counts[N] | offsets[N+1] | cursor[N] | bucket[E]
    unsigned* counts  = (unsigned*)d_ws;
    unsigned* offsets = counts + N;
    unsigned* cursor  = offsets + N + 1;
    unsigned* bucket  = cursor + N;

    pp_zero   <<<(N + 255) / 256, 256, 0, stream>>>(counts, N);
    pp_count  <<<(E + 255) / 256, 256, 0, stream>>>(ei, counts, E);
    pp_scan   <<<1, 1024, 0, stream>>>(counts, offsets, cursor, N);
    pp_scatter<<<(E + 255) / 256, 256, 0, stream>>>(ei, cursor, bucket, E);
    pn_reduce <<<(N + 7) / 8, 256, 0, stream>>>(x, pos, offsets, bucket, out, N);
}